// PositionEncoder_80126909874709
// MI455X (gfx1250) — compile-verified
//
#include <hip/hip_runtime.h>
#include <math.h>

#define NNODE 2048
#define NEDGE 8192
#define NMOL  128
#define NDIM  128
#define EDIM  64
#define PQ    32
#define H2    64
#define G4    256
#define FEATK 160
#define UH    128
#define TAUF  0.25f
#define NLAYERS 4

typedef float v2f __attribute__((ext_vector_type(2)));
typedef float v8f __attribute__((ext_vector_type(8)));

__device__ __forceinline__ float sigf(float x) { return 1.0f / (1.0f + expf(-x)); }

__device__ __forceinline__ float wred(float v) {
#pragma unroll
  for (int m = 16; m > 0; m >>= 1) v += __shfl_xor(v, m, 32);
  return v;
}

// ---------------------------------------------------------------- init
__global__ void k_init(float* s, float* c) { *s = 0.f; *c = 0.f; }

// ---------------------------------------------------------------- edge weights
__global__ __launch_bounds__(256)
void k_edge_w(const float* __restrict__ vf, const float* __restrict__ ef,
              const int* __restrict__ us, const int* __restrict__ vs,
              const float* __restrict__ We, const float* __restrict__ be,
              float* __restrict__ wout) {
  int wid = threadIdx.x >> 5, lane = threadIdx.x & 31;
  int j = blockIdx.x * 8 + wid;
  if (j >= NEDGE) return;
  int u = us[j], v = vs[j];
  float s = 0.f;
  for (int k = lane; k < 2 * NDIM + EDIM; k += 32) {
    float x;
    if (k < NDIM)            x = vf[u * NDIM + k];
    else if (k < NDIM + EDIM) x = ef[j * EDIM + (k - NDIM)];
    else                      x = vf[v * NDIM + (k - NDIM - EDIM)];
    s += x * We[k];
  }
  s = wred(s);
  if (lane == 0) wout[j] = sigf(s + be[0]);
}

// ---------------------------------------------------------------- adjacency build (deterministic)
__global__ __launch_bounds__(256)
void k_cnt_deg(const int* __restrict__ us, const int* __restrict__ vs,
               const float* __restrict__ w, int* __restrict__ cnt,
               float* __restrict__ deg) {
  int n = blockIdx.x * blockDim.x + threadIdx.x;
  if (n >= NNODE) return;
  int c = 0; float d = 0.f;
  for (int j = 0; j < NEDGE; ++j) {
    int u = us[j], v = vs[j];
    if (u == n || v == n) {
      ++c;
      float wj = w[j];
      if (u == v) d += wj;
      else {
        if (u == n) d += 2.f * wj;
        if (v == n) d += 2.f * wj;
      }
    }
  }
  cnt[n] = c; deg[n] = d;
}

__global__ void k_scan(const int* cnt, int* off) {
  if (threadIdx.x == 0 && blockIdx.x == 0) {
    int a = 0;
    for (int n = 0; n < NNODE; ++n) { off[n] = a; a += cnt[n]; }
  }
}

__global__ __launch_bounds__(256)
void k_fill(const int* __restrict__ us, const int* __restrict__ vs,
            const int* __restrict__ off, int* __restrict__ idx) {
  int n = blockIdx.x * blockDim.x + threadIdx.x;
  if (n >= NNODE) return;
  int p = off[n];
  for (int j = 0; j < NEDGE; ++j) {
    int u = us[j], v = vs[j];
    if (u == n || v == n) idx[p++] = j;
  }
}

// ---------------------------------------------------------------- WMMA f32 GEMM
// C(M,N) = A(M,K) @ B(K,N) + bias[N].  grid = (M/16, N/64), block = 32 (one wave).
// K must be a multiple of 32. Uses V_WMMA_F32_16X16X4_F32 (exact f32 path).
__global__ __launch_bounds__(32)
void k_wmma_gemm(const float* __restrict__ A, const float* __restrict__ B,
                 const float* __restrict__ bias, float* __restrict__ C,
                 int K, int N) {
  __shared__ float As[16][33];
  const int r0 = blockIdx.x * 16;
  const int c0 = blockIdx.y * 64;
  const int lane = threadIdx.x;
  const int arow = lane & 15;
  const int ksel = (lane >> 4) << 1;   // lanes 0-15 -> k+0/k+1, lanes 16-31 -> k+2/k+3
  v8f acc0 = {}, acc1 = {}, acc2 = {}, acc3 = {};

  for (int kc = 0; kc < K; kc += 32) {
#pragma unroll
    for (int i = 0; i < 16; ++i)
      As[i][lane] = A[(size_t)(r0 + i) * K + kc + lane];   // coalesced
    __syncthreads();
#pragma unroll
    for (int kk = 0; kk < 32; kk += 4) {
      int kb = kk + ksel;
      v2f a; a.x = As[arow][kb]; a.y = As[arow][kb + 1];
      const float* B0 = B + (size_t)(kc + kb) * N;
      const float* B1 = B + (size_t)(kc + kb + 1) * N;
      int bc = c0 + arow;
      v2f b0; b0.x = B0[bc];      b0.y = B1[bc];
      v2f b1; b1.x = B0[bc + 16]; b1.y = B1[bc + 16];
      v2f b2; b2.x = B0[bc + 32]; b2.y = B1[bc + 32];
      v2f b3; b3.x = B0[bc + 48]; b3.y = B1[bc + 48];
      acc0 = __builtin_amdgcn_wmma_f32_16x16x4_f32(false, a, false, b0, (short)0, acc0, false, false);
      acc1 = __builtin_amdgcn_wmma_f32_16x16x4_f32(false, a, false, b1, (short)0, acc1, false, false);
      acc2 = __builtin_amdgcn_wmma_f32_16x16x4_f32(false, a, false, b2, (short)0, acc2, false, false);
      acc3 = __builtin_amdgcn_wmma_f32_16x16x4_f32(false, a, false, b3, (short)0, acc3, false, false);
    }
    __syncthreads();
  }
  const int rbase = r0 + ((lane >> 4) << 3);
  const int cb = c0 + arow;
  float bi0 = bias ? bias[cb]      : 0.f;
  float bi1 = bias ? bias[cb + 16] : 0.f;
  float bi2 = bias ? bias[cb + 32] : 0.f;
  float bi3 = bias ? bias[cb + 48] : 0.f;
#pragma unroll
  for (int j = 0; j < 8; ++j) {
    C[(size_t)(rbase + j) * N + cb]      = acc0[j] + bi0;
    C[(size_t)(rbase + j) * N + cb + 16] = acc1[j] + bi1;
    C[(size_t)(rbase + j) * N + cb + 32] = acc2[j] + bi2;
    C[(size_t)(rbase + j) * N + cb + 48] = acc3[j] + bi3;
  }
}

// ---------------------------------------------------------------- sequential LSTM
__global__ __launch_bounds__(256)
void k_lstm(const float* __restrict__ xg, const float* __restrict__ Whh,
            float* __restrict__ p, float* __restrict__ q) {
  __shared__ float hsh[H2];
  __shared__ float gsh[G4];
  int col = threadIdx.x;
  float regW[H2];
#pragma unroll
  for (int k = 0; k < H2; ++k) regW[k] = Whh[k * G4 + col];
  if (col < H2) hsh[col] = 0.f;
  float c = 0.f;
  __syncthreads();
  for (int t = 0; t < NNODE; ++t) {
    float g = xg[t * G4 + col];           // bias already folded in
#pragma unroll
    for (int k = 0; k < H2; ++k) g += hsh[k] * regW[k];
    gsh[col] = g;
    __syncthreads();
    if (col < H2) {
      float iv = sigf(gsh[col]);
      float fv = sigf(gsh[col + H2]);
      float gv = tanhf(gsh[col + 2 * H2]);
      float ov = sigf(gsh[col + 3 * H2]);
      c = fv * c + iv * gv;
      float h = ov * tanhf(c);
      hsh[col] = h;
      if (col < PQ) p[t * PQ + col] = h;
      else          q[t * PQ + (col - PQ)] = h;
    }
    __syncthreads();
  }
}

// ---------------------------------------------------------------- per-molecule mean, subtract
__global__ __launch_bounds__(32)
void k_center_mean(const float* __restrict__ Mm, const float* __restrict__ q,
                   float* __restrict__ mean) {
  int m = blockIdx.x, d = threadIdx.x;
  float s = 0.f, cnt = 0.f;
  for (int n = 0; n < NNODE; ++n) {
    float mv = Mm[m * NNODE + n];
    cnt += mv;
    s += mv * q[n * PQ + d];
  }
  mean[m * PQ + d] = s / fmaxf(cnt, 1.0f);
}

__global__ __launch_bounds__(256)
void k_center_sub(const float* __restrict__ Mm, const float* __restrict__ mean,
                  float* __restrict__ q) {
  int t = blockIdx.x * blockDim.x + threadIdx.x;
  if (t >= NNODE * PQ) return;
  int n = t >> 5, d = t & 31;
  float s = 0.f;
  for (int m = 0; m < NMOL; ++m) s += Mm[m * NNODE + n] * mean[m * PQ + d];
  q[t] -= s;
}

// ---------------------------------------------------------------- mass / lambda
__global__ __launch_bounds__(256)
void k_mass_lam(const float* __restrict__ vf, const float* __restrict__ Wm,
                const float* __restrict__ bm, const float* __restrict__ Wd,
                const float* __restrict__ bd, float* __restrict__ mass,
                float* __restrict__ lam) {
  int wid = threadIdx.x >> 5, lane = threadIdx.x & 31;
  int n = blockIdx.x * 8 + wid;
  if (n >= NNODE) return;
  float sm = 0.f, sd = 0.f;
  for (int k = lane; k < NDIM; k += 32) {
    float x = vf[n * NDIM + k];
    sm += x * Wm[k]; sd += x * Wd[k];
  }
  sm = wred(sm); sd = wred(sd);
  if (lane == 0) {
    float xm = sm + bm[0];
    mass[n] = fmaxf(xm, 0.f) + log1pf(expf(-fabsf(xm)));   // stable softplus
    lam[n]  = sigf(sd + bd[0]);
  }
}

// ---------------------------------------------------------------- feat[:, :128] = v_features
__global__ __launch_bounds__(256)
void k_copy_featv(const float* __restrict__ vf, float* __restrict__ feat) {
  int t = blockIdx.x * blockDim.x + threadIdx.x;
  if (t >= NNODE * NDIM) return;
  int n = t >> 7, k = t & 127;
  feat[n * FEATK + k] = vf[t];
}

// ---------------------------------------------------------------- per-layer prep: feat q cols + |q|^2
__global__ __launch_bounds__(256)
void k_prep(const float* __restrict__ q, float* __restrict__ feat,
            float* __restrict__ sqn) {
  int wid = threadIdx.x >> 5, lane = threadIdx.x & 31;
  int n = blockIdx.x * 8 + wid;
  if (n >= NNODE) return;
  float qv = q[n * PQ + lane];
  feat[n * FEATK + NDIM + lane] = qv;
  float s = wred(qv * qv);
  if (lane == 0) sqn[n] = s;
}

// ---------------------------------------------------------------- u-net value + q-gradient
__global__ __launch_bounds__(128)
void k_ugrad(const float* __restrict__ z, const float* __restrict__ Wu1,
             const float* __restrict__ Wu2, const float* __restrict__ bu2,
             float* __restrict__ uval, float* __restrict__ qgu) {
  __shared__ float tsh[UH];
  __shared__ float rsh[UH];
  int n = blockIdx.x, j = threadIdx.x;
  float zj = z[n * UH + j];
  float w2 = Wu2[j];
  bool on = zj > 0.f;
  tsh[j] = on ? w2 : 0.f;
  rsh[j] = on ? zj * w2 : 0.f;
  __syncthreads();
  for (int s = 64; s > 0; s >>= 1) { if (j < s) rsh[j] += rsh[j + s]; __syncthreads(); }
  if (j == 0) uval[n] = rsh[0] + bu2[0];
  if (j < PQ) {
    float s = 0.f;
    for (int i = 0; i < UH; ++i) s += tsh[i] * Wu1[(NDIM + j) * UH + i];
    qgu[n * PQ + j] = s;
  }
}

// ---------------------------------------------------------------- sparse e@q and e@|q|^2 (deterministic gather)
__global__ __launch_bounds__(256)
void k_gather(const int* __restrict__ us, const int* __restrict__ vs,
              const float* __restrict__ w, const int* __restrict__ cnt,
              const int* __restrict__ off, const int* __restrict__ idx,
              const float* __restrict__ q, const float* __restrict__ sqn,
              float* __restrict__ eq, float* __restrict__ esq) {
  int wid = threadIdx.x >> 5, lane = threadIdx.x & 31;
  int n = blockIdx.x * 8 + wid;
  if (n >= NNODE) return;
  int c = cnt[n], o = off[n];
  float acc = 0.f, eacc = 0.f;
  for (int i = 0; i < c; ++i) {
    int j = idx[o + i];
    int u = us[j], v = vs[j];
    float wj = w[j];
    if (u == v) { acc += wj * q[u * PQ + lane]; eacc += wj * sqn[u]; }
    else {
      acc  += wj * (q[u * PQ + lane] + q[v * PQ + lane]);
      eacc += wj * (sqn[u] + sqn[v]);
    }
  }
  eq[n * PQ + lane] = acc;
  if (lane == 0) esq[n] = eacc;
}

// ---------------------------------------------------------------- node energy/grad/update
__global__ __launch_bounds__(256)
void k_node(const float* __restrict__ deg, const float* __restrict__ mass,
            const float* __restrict__ lam, const float* __restrict__ eq,
            const float* __restrict__ esq, const float* __restrict__ sqn,
            const float* __restrict__ uval, const float* __restrict__ qgu,
            float* __restrict__ p, float* __restrict__ q,
            float* __restrict__ dpb, float* __restrict__ ssum,
            float* __restrict__ en, float* __restrict__ pe) {
  int wid = threadIdx.x >> 5, lane = threadIdx.x & 31;
  int n = blockIdx.x * 8 + wid;
  if (n >= NNODE) return;
  float pd  = p[n * PQ + lane], qd = q[n * PQ + lane];
  float eqd = eq[n * PQ + lane];
  float mn = mass[n], dn = deg[n], ln = lam[n];
  float gp = mn * pd;
  float gq = 2.f * dn * qd - 2.f * eqd + qgu[n * PQ + lane];
  float dq = gp;
  float dp = -gq - ln * pd;
  float sd = dq - pd;
  float s2 = wred(sd * sd);
  float ps = wred(pd * pd);
  float qe = wred(qd * eqd);
  if (lane == 0) {
    ssum[n] = s2;
    en[n] = 0.5f * mn * ps + 0.5f * (dn * sqn[n] - 2.f * qe + esq[n]) + uval[n];
    pe[n] = 0.5f * ln * ps;
  }
  dpb[n * PQ + lane] = dp;
  p[n * PQ + lane] = pd + TAUF * dp;
  q[n * PQ + lane] = qd + TAUF * dq;
}

// ---------------------------------------------------------------- per-molecule reductions
__global__ __launch_bounds__(32)
void k_mol(const float* __restrict__ Mm, const float* __restrict__ dpb,
           const float* __restrict__ en, const float* __restrict__ pe,
           float* __restrict__ cpart, float* __restrict__ hmol,
           float* __restrict__ dmol, int last) {
  int m = blockIdx.x, lane = threadIdx.x;
  float cm = 0.f, hm = 0.f, dm = 0.f;
  for (int n = 0; n < NNODE; ++n) {
    float mv = Mm[m * NNODE + n];
    cm += mv * dpb[n * PQ + lane];
    if (lane == 0) { hm += mv * en[n]; dm += mv * pe[n]; }
  }
  cm *= TAUF;
  float cs = wred(cm * cm);
  if (lane == 0) {
    cpart[m] = cs;
    if (last) { hmol[m] = hm; dmol[m] = dm; }
  }
}

// ---------------------------------------------------------------- deterministic loss accumulation
__global__ __launch_bounds__(256)
void k_finalize(const float* __restrict__ ssum, const float* __restrict__ cpart,
                float* __restrict__ sout, float* __restrict__ cout) {
  __shared__ float sh[256];
  int t = threadIdx.x;
  float a = 0.f;
  for (int i = t; i < NNODE; i += 256) a += ssum[i];
  sh[t] = a; __syncthreads();
  for (int s = 128; s > 0; s >>= 1) { if (t < s) sh[t] += sh[t + s]; __syncthreads(); }
  if (t == 0) *sout += sqrtf(sh[0]);
  __syncthreads();
  sh[t] = (t < NMOL) ? cpart[t] : 0.f;
  __syncthreads();
  for (int s = 128; s > 0; s >>= 1) { if (t < s) sh[t] += sh[t + s]; __syncthreads(); }
  if (t == 0) *cout += sqrtf(sh[0]);
}

// ================================================================ launcher
extern "C" void kernel_launch(void* const* d_in, const int* in_sizes, int n_in,
                              void* d_out, int out_size, void* d_ws, size_t ws_size,
                              hipStream_t stream) {
  const float* vf   = (const float*)d_in[0];
  const float* ef   = (const float*)d_in[1];
  const int*   us   = (const int*)d_in[2];
  const int*   vs   = (const int*)d_in[3];
  const float* Mm   = (const float*)d_in[4];
  const float* We   = (const float*)d_in[8];
  const float* be   = (const float*)d_in[9];
  const float* Wih  = (const float*)d_in[10];
  const float* Whh  = (const float*)d_in[11];
  const float* blst = (const float*)d_in[12];
  const float* Wm   = (const float*)d_in[13];
  const float* bm   = (const float*)d_in[14];
  const float* Wd   = (const float*)d_in[15];
  const float* bd   = (const float*)d_in[16];
  const float* Wu1  = (const float*)d_in[17];
  const float* bu1  = (const float*)d_in[18];
  const float* Wu2  = (const float*)d_in[19];
  const float* bu2  = (const float*)d_in[20];

  float* out  = (float*)d_out;
  float* p    = out;
  float* q    = out + NNODE * PQ;
  float* sout = out + 2 * NNODE * PQ;
  float* cout = sout + 1;
  float* hmol = cout + 1;
  float* dmol = hmol + NMOL;

  float* ws   = (float*)d_ws;
  float* wE   = ws;                       // 8192
  float* deg  = wE + NEDGE;               // 2048
  float* xg   = deg + NNODE;              // 2048*256
  float* feat = xg + NNODE * G4;          // 2048*160
  float* zbuf = feat + NNODE * FEATK;     // 2048*128
  float* uval = zbuf + NNODE * UH;        // 2048
  float* qgu  = uval + NNODE;             // 2048*32
  float* eqb  = qgu + NNODE * PQ;         // 2048*32
  float* esq  = eqb + NNODE * PQ;         // 2048
  float* sqn  = esq + NNODE;              // 2048
  float* mass = sqn + NNODE;              // 2048
  float* lam  = mass + NNODE;             // 2048
  float* en   = lam + NNODE;              // 2048
  float* pe   = en + NNODE;               // 2048
  float* dpb  = pe + NNODE;               // 2048*32
  float* ssum = dpb + NNODE * PQ;         // 2048
  float* cpart = ssum + NNODE;            // 128
  float* mean = cpart + NMOL;             // 128*32
  int* cnt  = (int*)(mean + NMOL * PQ);   // 2048
  int* off  = cnt + NNODE;                // 2048
  int* aidx = off + NNODE;                // 16384

  k_init<<<1, 1, 0, stream>>>(sout, cout);
  k_edge_w<<<NEDGE / 8, 256, 0, stream>>>(vf, ef, us, vs, We, be, wE);
  k_cnt_deg<<<NNODE / 256, 256, 0, stream>>>(us, vs, wE, cnt, deg);
  k_scan<<<1, 1, 0, stream>>>(cnt, off);
  k_fill<<<NNODE / 256, 256, 0, stream>>>(us, vs, off, aidx);

  dim3 g1(NNODE / 16, G4 / 64);
  k_wmma_gemm<<<g1, 32, 0, stream>>>(vf, Wih, blst, xg, NDIM, G4);   // xg = v@W_ih + b_lstm
  k_lstm<<<1, 256, 0, stream>>>(xg, Whh, p, q);
  k_center_mean<<<NMOL, 32, 0, stream>>>(Mm, q, mean);
  k_center_sub<<<(NNODE * PQ) / 256, 256, 0, stream>>>(Mm, mean, q);
  k_mass_lam<<<NNODE / 8, 256, 0, stream>>>(vf, Wm, bm, Wd, bd, mass, lam);
  k_copy_featv<<<(NNODE * NDIM) / 256, 256, 0, stream>>>(vf, feat);

  for (int L = 0; L < NLAYERS; ++L) {
    k_prep<<<NNODE / 8, 256, 0, stream>>>(q, feat, sqn);
    dim3 g2(NNODE / 16, UH / 64);
    k_wmma_gemm<<<g2, 32, 0, stream>>>(feat, Wu1, bu1, zbuf, FEATK, UH);  // z = [v,q]@W_u1 + b_u1
    k_ugrad<<<NNODE, 128, 0, stream>>>(zbuf, Wu1, Wu2, bu2, uval, qgu);
    k_gather<<<NNODE / 8, 256, 0, stream>>>(us, vs, wE, cnt, off, aidx, q, sqn, eqb, esq);
    k_node<<<NNODE / 8, 256, 0, stream>>>(deg, mass, lam, eqb, esq, sqn, uval, qgu,
                                          p, q, dpb, ssum, en, pe);
    k_mol<<<NMOL, 32, 0, stream>>>(Mm, dpb, en, pe, cpart, hmol, dmol,
                                   (L == NLAYERS - 1) ? 1 : 0);
    k_finalize<<<1, 256, 0, stream>>>(ssum, cpart, sout, cout);
  }
}